// GATLayer_7121055777013
// MI455X (gfx1250) — compile-verified
//
#include <hip/hip_runtime.h>

#define IN_C  256
#define OUT_C 256
#define NEG_SLOPE 0.2f

typedef __attribute__((ext_vector_type(2))) float v2f;
typedef __attribute__((ext_vector_type(8))) float v8f;

// ---------- float <-> order-preserving unsigned encoding (for atomic segment-max) ----------
__device__ __forceinline__ unsigned f32_order(float f) {
    unsigned u = __float_as_uint(f);
    return (u & 0x80000000u) ? ~u : (u | 0x80000000u);
}
__device__ __forceinline__ float order_f32(unsigned u) {
    return __uint_as_float((u & 0x80000000u) ? (u ^ 0x80000000u) : ~u);
}

// ---------- kernel 1a: init per-node accumulators ----------
__global__ __launch_bounds__(256) void init_nodes(unsigned* __restrict__ emax_u,
                                                  float* __restrict__ denom, int N) {
    int i = blockIdx.x * blockDim.x + threadIdx.x;
    if (i < N) { emax_u[i] = 0u; denom[i] = 0.0f; }
}

// ---------- kernel 1b: out[i][c] = bias[c]  (atomic accumulation target) ----------
__global__ __launch_bounds__(256) void init_out(float* __restrict__ out,
                                                const float* __restrict__ bias, int total) {
    int i = blockIdx.x * blockDim.x + threadIdx.x;
    if (i < total) out[i] = bias[i & (OUT_C - 1)];
}

// ---------- kernel 2: h = x @ W via V_WMMA_F32_16X16X4_F32 ----------
// grid.x = N/16 row tiles; 512 threads = 16 waves; wave w computes 16x16 tile at cols [16w,16w+16).
// A stripe (16x256 of x) is staged into LDS with CDNA5 async copies
// (GLOBAL_LOAD_ASYNC_TO_LDS_B128, tracked by ASYNCcnt) — no VGPR round trip.
#define AS_STRIDE (IN_C + 4)   // pad so the 16 A-rows hit distinct LDS banks (1040 B, 16B-aligned)
__global__ __launch_bounds__(512) void gemm_h(const float* __restrict__ x,
                                              const float* __restrict__ W,
                                              float* __restrict__ h, int N) {
    __shared__ __align__(16) float As[16 * AS_STRIDE];
    const int row0 = blockIdx.x * 16;
    const float* stripe = x + (size_t)row0 * IN_C;   // SGPR base for GVS-mode async copy

    // async-stage the 16x256 stripe: 1024 x b128, 2 per thread
    for (int t = threadIdx.x; t < 16 * (IN_C / 4); t += blockDim.x) {
        int r  = t >> 6;                      // 64 float4 per row
        int c4 = t & 63;
        unsigned goff = (unsigned)(r * IN_C + c4 * 4) * 4u;          // global byte offset
        unsigned lds  = (unsigned)(uintptr_t)&As[r * AS_STRIDE + c4 * 4];  // LDS byte address
        asm volatile("global_load_async_to_lds_b128 %0, %1, %2"
                     :: "v"(lds), "v"(goff), "s"(stripe) : "memory");
    }
    asm volatile("s_wait_asynccnt 0x0" ::: "memory");  // my wave's copies landed in LDS
    __syncthreads();                                    // everyone's copies visible

    const int wave = threadIdx.x >> 5;      // 0..15 -> col tile
    const int lane = threadIdx.x & 31;
    const int col0 = wave * 16;
    const int m    = lane & 15;
    const bool hi  = lane >= 16;            // lanes 16..31 carry K+2/K+3 of A, K rows 2/3 of B

    v8f acc = {};
    for (int k0 = 0; k0 < IN_C; k0 += 4) {
        const int ka = k0 + (hi ? 2 : 0);
        v2f a;                               // A 16x4 f32: v0=K, v1=K+1 (lanes>=16: K+2,K+3)
        a.x = As[m * AS_STRIDE + ka];
        a.y = As[m * AS_STRIDE + ka + 1];
        v2f b;                               // B 4x16 f32: row K striped over lanes
        b.x = W[(size_t)ka * OUT_C + col0 + m];
        b.y = W[(size_t)(ka + 1) * OUT_C + col0 + m];
        acc = __builtin_amdgcn_wmma_f32_16x16x4_f32(
                  false, a, false, b, (short)0, acc, false, false);
    }

    // C/D layout: VGPR v -> M=v (lanes 0-15) / M=v+8 (lanes 16-31), N = lane&15
    const int n     = col0 + m;
    const int mbase = row0 + (hi ? 8 : 0);
#pragma unroll
    for (int v = 0; v < 8; ++v)
        h[(size_t)(mbase + v) * OUT_C + n] = acc[v];
}

// ---------- kernel 3: per-row attention dots  a_s = h.att_src, a_d = h.att_dst ----------
__global__ __launch_bounds__(256) void rowdot(const float* __restrict__ h,
                                              const float* __restrict__ att_src,
                                              const float* __restrict__ att_dst,
                                              float* __restrict__ a_s,
                                              float* __restrict__ a_d, int N) {
    int row = blockIdx.x * 8 + (threadIdx.x >> 5);
    if (row >= N) return;
    int lane = threadIdx.x & 31;
    const float4* hp = (const float4*)(h + (size_t)row * OUT_C);
    const float4* sp = (const float4*)att_src;
    const float4* dp = (const float4*)att_dst;
    float ss = 0.f, dd = 0.f;
#pragma unroll
    for (int i = lane; i < OUT_C / 4; i += 32) {
        float4 hv = hp[i], sv = sp[i], dv = dp[i];
        ss += hv.x * sv.x + hv.y * sv.y + hv.z * sv.z + hv.w * sv.w;
        dd += hv.x * dv.x + hv.y * dv.y + hv.z * dv.z + hv.w * dv.w;
    }
#pragma unroll
    for (int off = 16; off > 0; off >>= 1) {
        ss += __shfl_down(ss, off, 32);
        dd += __shfl_down(dd, off, 32);
    }
    if (lane == 0) { a_s[row] = ss; a_d[row] = dd; }
}

// ---------- kernel 4: e = leaky(a_s[src]+a_d[dst]); atomic segment-max ----------
__global__ __launch_bounds__(256) void edge_pass1(const int* __restrict__ ei,
                                                  const float* __restrict__ a_s,
                                                  const float* __restrict__ a_d,
                                                  float* __restrict__ e_edge,
                                                  unsigned* __restrict__ emax_u,
                                                  int E, int EA) {
    int idx = blockIdx.x * blockDim.x + threadIdx.x;
    if (idx >= EA) return;
    int s, d;
    if (idx < E) { s = ei[idx]; d = ei[E + idx]; }
    else         { s = d = idx - E; }                 // self loop
    float e = a_s[s] + a_d[d];
    e = e > 0.f ? e : NEG_SLOPE * e;
    e_edge[idx] = e;
    atomicMax(emax_u + d, f32_order(e));
}

// ---------- kernel 5: e_exp = exp(e - max[dst]); atomic denom ----------
__global__ __launch_bounds__(256) void edge_pass2(const int* __restrict__ ei,
                                                  const float* __restrict__ e_edge,
                                                  const unsigned* __restrict__ emax_u,
                                                  float* __restrict__ e_exp,
                                                  float* __restrict__ denom,
                                                  int E, int EA) {
    int idx = blockIdx.x * blockDim.x + threadIdx.x;
    if (idx >= EA) return;
    int d = (idx < E) ? ei[E + idx] : (idx - E);
    float ex = expf(e_edge[idx] - order_f32(emax_u[d]));
    e_exp[idx] = ex;
    atomicAdd(denom + d, ex);
}

// ---------- kernel 6: out[dst] += (e_exp/denom[dst]) * h[src] ; wave per edge ----------
// h (51 MB) is resident in the 192 MB L2, so the 1.7 GB gather and the f32
// atomic scatter are served from L2, not HBM.
__global__ __launch_bounds__(256) void scatter_out(const int* __restrict__ ei,
                                                   const float* __restrict__ h,
                                                   const float* __restrict__ e_exp,
                                                   const float* __restrict__ denom,
                                                   float* __restrict__ out,
                                                   int E, int EA) {
    int idx = blockIdx.x * 8 + (threadIdx.x >> 5);
    if (idx >= EA) return;
    int lane = threadIdx.x & 31;
    int s, d;
    if (idx < E) { s = ei[idx]; d = ei[E + idx]; }
    else         { s = d = idx - E; }
    float alpha = e_exp[idx] / denom[d];
    const float4* hp = (const float4*)(h + (size_t)s * OUT_C);
    float* op = out + (size_t)d * OUT_C;
#pragma unroll
    for (int half = 0; half < 2; ++half) {
        int c4 = half * 32 + lane;      // float4 index 0..63, coalesced 512B per half
        float4 hv = hp[c4];
        int c = c4 * 4;
        atomicAdd(op + c + 0, alpha * hv.x);
        atomicAdd(op + c + 1, alpha * hv.y);
        atomicAdd(op + c + 2, alpha * hv.z);
        atomicAdd(op + c + 3, alpha * hv.w);
    }
}

// ---------- host launcher ----------
extern "C" void kernel_launch(void* const* d_in, const int* in_sizes, int n_in,
                              void* d_out, int out_size, void* d_ws, size_t ws_size,
                              hipStream_t stream) {
    const float* x       = (const float*)d_in[0];
    const int*   ei      = (const int*)  d_in[1];
    const float* W       = (const float*)d_in[2];
    const float* att_src = (const float*)d_in[3];
    const float* att_dst = (const float*)d_in[4];
    const float* bias    = (const float*)d_in[5];
    float* out = (float*)d_out;

    const int N  = in_sizes[0] / IN_C;   // 50000
    const int E  = in_sizes[1] / 2;      // 1,600,000
    const int EA = E + N;                // edges + self loops

    // carve workspace (256B-aligned slices)
    auto align256 = [](size_t v) { return (v + 255) & ~(size_t)255; };
    char* ws = (char*)d_ws;
    size_t off = 0;
    float*    h      = (float*)(ws + off); off = align256(off + (size_t)N * OUT_C * sizeof(float));
    float*    a_s    = (float*)(ws + off); off = align256(off + (size_t)N * sizeof(float));
    float*    a_d    = (float*)(ws + off); off = align256(off + (size_t)N * sizeof(float));
    unsigned* emax_u = (unsigned*)(ws + off); off = align256(off + (size_t)N * sizeof(unsigned));
    float*    denom  = (float*)(ws + off); off = align256(off + (size_t)N * sizeof(float));
    float*    e_edge = (float*)(ws + off); off = align256(off + (size_t)EA * sizeof(float));
    float*    e_exp  = (float*)(ws + off); off = align256(off + (size_t)EA * sizeof(float));
    (void)ws_size; (void)n_in; (void)out_size;

    init_nodes<<<(N + 255) / 256, 256, 0, stream>>>(emax_u, denom, N);
    init_out<<<((size_t)N * OUT_C + 255) / 256, 256, 0, stream>>>(out, bias, N * OUT_C);
    gemm_h<<<N / 16, 512, 0, stream>>>(x, W, h, N);
    rowdot<<<(N + 7) / 8, 256, 0, stream>>>(h, att_src, att_dst, a_s, a_d, N);
    edge_pass1<<<(EA + 255) / 256, 256, 0, stream>>>(ei, a_s, a_d, e_edge, emax_u, E, EA);
    edge_pass2<<<(EA + 255) / 256, 256, 0, stream>>>(ei, e_edge, emax_u, e_exp, denom, E, EA);
    scatter_out<<<(EA + 7) / 8, 256, 0, stream>>>(ei, h, e_exp, denom, out, E, EA);
}